// TokenSelection_60868276518924
// MI455X (gfx1250) — compile-verified
//
#include <hip/hip_runtime.h>
#include <hip/hip_bf16.h>

// Problem constants (from reference)
#define B        32
#define S        8192
#define D        512
#define BLK      64          // BLOCK_SIZE
#define NBLK     128         // S / BLK
#define KSEL     16          // NUM_SELECTED
#define TPB      256
#define CHUNKS   4           // float4 chunks per thread

#if defined(__has_builtin)
#if __has_builtin(__builtin_amdgcn_global_load_async_to_lds_b128) && \
    __has_builtin(__builtin_amdgcn_global_store_async_from_lds_b128)
#define USE_ASYNC_LDS 1
#endif
#endif

// Builtin prototype (from hipcc diagnostic): params are 4xi32-vector pointers,
// global side in AS1, LDS side in AS3.
typedef int v4i __attribute__((vector_size(4 * sizeof(int))));
typedef __attribute__((address_space(1))) void  gbl_void_t;
typedef __attribute__((address_space(3))) void  lds_void_t;
typedef __attribute__((address_space(1))) v4i*  gbl_v4i_p;
typedef __attribute__((address_space(3))) v4i*  lds_v4i_p;

__device__ __forceinline__ void wait_async0() {
#if __has_builtin(__builtin_amdgcn_s_wait_asynccnt)
    __builtin_amdgcn_s_wait_asynccnt(0);
#else
    asm volatile("s_wait_asynccnt 0" ::: "memory");
#endif
}

// -------- Kernel: HBM-bound gather of selected key blocks -----------------
// out[b, k*64 + t, :] = keys[b, blk[b][k]*64 + t, :]
// Each thread moves 4 x 16B, staged through LDS with the CDNA5 async
// global<->LDS DMA engine (ASYNCcnt). Pipeline: 4 async loads -> one
// s_wait_asynccnt 0 -> 4 async stores. Per-wave each async op covers a
// contiguous 512B (lanes consecutive, 16B each). S_ENDPGM's implicit
// wait-idle covers the outstanding stores.
__global__ void gather_blocks_kernel(const float* __restrict__ keys,
                                     const int* __restrict__ blk_idx,
                                     float* __restrict__ out) {
    __shared__ __align__(16) char stage[TPB * CHUNKS * 16];

    const int tid = threadIdx.x;
    const long long base = (long long)blockIdx.x * (TPB * CHUNKS);

#ifdef USE_ASYNC_LDS
    float* dsts[CHUNKS];
#pragma unroll
    for (int j = 0; j < CHUNKS; ++j) {
        long long i  = base + j * TPB + tid;    // flat float4 index
        int d4       = (int)(i & (D / 4 - 1));  // float4 within row (0..127)
        long long rw = i >> 7;                  // output row
        int t        = (int)(rw & (BLK - 1));   // token within block
        int k        = (int)((rw >> 6) & (KSEL - 1));
        int b        = (int)(rw >> 10);
        int blk      = blk_idx[b * KSEL + k];

        const float* src = keys + (((long long)b * S + (long long)blk * BLK + t) * D
                                   + (long long)d4 * 4);
        dsts[j] = out + i * 4;

        lds_v4i_p lptr = (lds_v4i_p)(lds_void_t*)(stage + (j * TPB + tid) * 16);
        __builtin_amdgcn_global_load_async_to_lds_b128(
            (gbl_v4i_p)(gbl_void_t*)src, lptr, 0, 0);
    }
    wait_async0();  // all 4 loads landed in LDS
#pragma unroll
    for (int j = 0; j < CHUNKS; ++j) {
        lds_v4i_p lptr = (lds_v4i_p)(lds_void_t*)(stage + (j * TPB + tid) * 16);
        __builtin_amdgcn_global_store_async_from_lds_b128(
            (gbl_v4i_p)(gbl_void_t*)dsts[j], lptr, 0, 0);
    }
#else
#pragma unroll
    for (int j = 0; j < CHUNKS; ++j) {
        long long i  = base + j * TPB + tid;
        int d4       = (int)(i & (D / 4 - 1));
        long long rw = i >> 7;
        int t        = (int)(rw & (BLK - 1));
        int k        = (int)((rw >> 6) & (KSEL - 1));
        int b        = (int)(rw >> 10);
        int blk      = blk_idx[b * KSEL + k];
        const float* src = keys + (((long long)b * S + (long long)blk * BLK + t) * D
                                   + (long long)d4 * 4);
        *(float4*)(out + i * 4) = *(const float4*)src;
    }
#endif
}

// -------- Kernel: per-batch top-16 of 128 block scores --------------------
// One lane per batch (32 lanes = 1 wave). Selection of 16 maxima with strict
// '>' comparison -> descending order, lowest index wins ties (matches
// jax.lax.top_k / torch.topk semantics). 32*16*128 = 64K compares: noise.
__global__ void topk_blocks_kernel(const float* __restrict__ scores,
                                   int* __restrict__ blk_idx) {
    int b = threadIdx.x;
    if (b >= B) return;
    const float* s = scores + (size_t)b * NBLK;
    unsigned mask[NBLK / 32] = {0u, 0u, 0u, 0u};
    for (int k = 0; k < KSEL; ++k) {
        float best = -__builtin_inff();
        int   bi   = 0;
        for (int j = 0; j < NBLK; ++j) {
            if (!((mask[j >> 5] >> (j & 31)) & 1u)) {
                float v = s[j];
                if (v > best) { best = v; bi = j; }
            }
        }
        mask[bi >> 5] |= 1u << (bi & 31);
        blk_idx[b * KSEL + k] = bi;
    }
}

extern "C" void kernel_launch(void* const* d_in, const int* in_sizes, int n_in,
                              void* d_out, int out_size, void* d_ws, size_t ws_size,
                              hipStream_t stream) {
    const float* keys   = (const float*)d_in[0];   // [32, 8192, 512] f32
    const float* scores = (const float*)d_in[1];   // [32, 128] f32
    float*       out    = (float*)d_out;           // [32, 1024, 512] f32
    int*         blk    = (int*)d_ws;              // [32, 16] selected block ids

    topk_blocks_kernel<<<1, 32, 0, stream>>>(scores, blk);

    // total float4s = 32 * 1024 * 128 = 4,194,304 -> 4096 blocks of 256x4
    const int nblocks = (B * KSEL * BLK * (D / 4)) / (TPB * CHUNKS);
    gather_blocks_kernel<<<nblocks, TPB, 0, stream>>>(keys, blk, out);
}